// TopKMoE_6597069767522
// MI455X (gfx1250) — compile-verified
//
#include <hip/hip_runtime.h>
#include <hip/hip_bf16.h>

// ---------------------------------------------------------------------------
// Top-K (2 of 8) MoE forward for MI455X / gfx1250, fp32 end-to-end.
//   T = 4096 tokens, d = 256, H = 512, E = 8, K = 2
// Routing-aware: only selected experts are computed (exactly matches the
// dense-masked reference math). Matrix work on V_WMMA_F32_16X16X4_F32.
// ---------------------------------------------------------------------------

typedef float v2f __attribute__((ext_vector_type(2)));
typedef float v8f __attribute__((ext_vector_type(8)));

constexpr int T_TOK = 4096;   // SEQ_LEN * BATCH
constexpr int DM    = 256;    // d_model
constexpr int HH    = 512;    // expert hidden dim
constexpr int NE    = 8;      // num experts
constexpr int XS_STRIDE = 260; // 256 + 4  -> conflict-free A fetches
constexpr int HS_STRIDE = 516; // 512 + 4

// ---------------------------------------------------------------------------
// Kernel 1: zero output (atomic-add target) and per-expert counters
// ---------------------------------------------------------------------------
__global__ void moe_zero(float* __restrict__ out, int n, int* __restrict__ counts) {
    int i = blockIdx.x * blockDim.x + threadIdx.x;
    if (i < n) out[i] = 0.0f;
    if (blockIdx.x == 0 && threadIdx.x < NE) counts[threadIdx.x] = 0;
}

// ---------------------------------------------------------------------------
// Kernel 2: gating. One wave32 per token. Exact fp32 logits, top-2, stable
// 2-way softmax, atomic compaction into per-expert (token, weight) lists.
// ---------------------------------------------------------------------------
__global__ __launch_bounds__(128) void moe_gate(
    const float* __restrict__ x,   const float* __restrict__ Wg,
    const float* __restrict__ bg,  const float* __restrict__ bias,
    int* __restrict__ counts, int* __restrict__ tokL, float* __restrict__ wtL)
{
    const int lane = threadIdx.x & 31;
    const int wv   = threadIdx.x >> 5;
    const int t    = blockIdx.x * 4 + wv;          // token id, grid = T/4

    float acc[NE];
#pragma unroll
    for (int e = 0; e < NE; ++e) acc[e] = 0.0f;

    // lane-parallel partial dot products for all 8 experts
    for (int d = lane; d < DM; d += 32) {
        float xv = x[(size_t)t * DM + d];
#pragma unroll
        for (int e = 0; e < NE; ++e) acc[e] += xv * Wg[d * NE + e];
    }
    // wave32 tree reduction
#pragma unroll
    for (int off = 16; off > 0; off >>= 1) {
#pragma unroll
        for (int e = 0; e < NE; ++e) acc[e] += __shfl_down(acc[e], off, 32);
    }

    if (lane == 0) {
        float lg[NE];
#pragma unroll
        for (int e = 0; e < NE; ++e) lg[e] = acc[e] + bg[e] + bias[e];
        // top-1 (first occurrence on ties, matching jax.lax.top_k)
        int i0 = 0; float v0 = lg[0];
#pragma unroll
        for (int e = 1; e < NE; ++e) if (lg[e] > v0) { v0 = lg[e]; i0 = e; }
        // top-2
        int i1 = (i0 == 0) ? 1 : 0; float v1 = lg[i1];
#pragma unroll
        for (int e = 0; e < NE; ++e)
            if (e != i0 && e != i1 && lg[e] > v1) { v1 = lg[e]; i1 = e; }
        // softmax over [v0, v1], v0 >= v1 -> stable
        float w1 = __expf(v1 - v0);
        float s  = 1.0f + w1;
        float w0 = 1.0f / s;  w1 = w1 / s;

        int p0 = atomicAdd(&counts[i0], 1);
        tokL[i0 * T_TOK + p0] = t;  wtL[i0 * T_TOK + p0] = w0;
        int p1 = atomicAdd(&counts[i1], 1);
        tokL[i1 * T_TOK + p1] = t;  wtL[i1 * T_TOK + p1] = w1;
    }
}

// ---------------------------------------------------------------------------
// Kernel 3: per-expert FFN on 16-token tiles with V_WMMA_F32_16X16X4_F32.
// Block = 256 threads (8 wave32s). grid = (T/16 tiles, E experts).
//   GEMM1: H = relu(X[16x256] @ W1e[256x512] + b1)   (4 col-tiles / wave)
//   GEMM2: Y = H[16x512] @ W2e[512x256] + b2         (2 col-tiles / wave)
//   scatter: out[token] += weight * Y_row (global_atomic_add_f32)
// f32 WMMA operand layout (ISA 7.12.2): A 16x4: lanes 0-15 K={0,1},
// lanes 16-31 K={2,3}; C/D: N = lane%16, M = vgpr + 8*(lane/16).
// ---------------------------------------------------------------------------
__global__ __launch_bounds__(256) void moe_ffn(
    const float* __restrict__ x,
    const float* __restrict__ W1, const float* __restrict__ b1,
    const float* __restrict__ W2, const float* __restrict__ b2,
    const int* __restrict__ counts, const int* __restrict__ tokL,
    const float* __restrict__ wtL, float* __restrict__ out)
{
    __shared__ float Xs[16 * XS_STRIDE];
    __shared__ float Hs[16 * HS_STRIDE];
    __shared__ int   toks[16];
    __shared__ float wts[16];

    const int e   = blockIdx.y;
    const int cnt = counts[e];
    const int m0  = blockIdx.x * 16;
    if (m0 >= cnt) return;                       // block-uniform: EXEC stays all-1s

    const int tid  = threadIdx.x;
    const int lane = tid & 31;
    const int wv   = tid >> 5;
    const int sub  = lane & 15;                  // N (or M for A-operand)
    const int hi   = lane >> 4;                  // selects K pair / M half

    const float* W1e = W1 + (size_t)e * DM * HH;
    const float* W2e = W2 + (size_t)e * HH * DM;

    // warm the L2/WGP$ for this expert's weight panels (global_prefetch_b8)
    __builtin_prefetch(W1e + (size_t)tid * HH, 0, 1);
    __builtin_prefetch(W2e + (size_t)tid * 2 * DM, 0, 1);

    if (tid < 16) {
        int m = m0 + tid;
        if (m < cnt) { toks[tid] = tokL[e * T_TOK + m]; wts[tid] = wtL[e * T_TOK + m]; }
        else         { toks[tid] = tokL[e * T_TOK + m0]; wts[tid] = 0.0f; }  // padded row
    }
    __syncthreads();

    // stage 16 token rows into LDS (coalesced: 16 floats per thread)
#pragma unroll
    for (int i = 0; i < 16; ++i)
        Xs[i * XS_STRIDE + tid] = x[(size_t)toks[i] * DM + tid];
    __syncthreads();

    // ---- GEMM1: H = relu(X @ W1e + b1) ------------------------------------
#pragma unroll
    for (int nt = 0; nt < 4; ++nt) {
        const int n0 = (wv * 4 + nt) * 16;
        v8f acc = {};
        for (int k0 = 0; k0 < DM; k0 += 4) {
            const int ka = k0 + 2 * hi;
            v2f a, b;
            a.x = Xs[sub * XS_STRIDE + ka];
            a.y = Xs[sub * XS_STRIDE + ka + 1];
            b.x = W1e[(size_t)ka * HH + n0 + sub];
            b.y = W1e[(size_t)(ka + 1) * HH + n0 + sub];
            acc = __builtin_amdgcn_wmma_f32_16x16x4_f32(
                false, a, false, b, (short)0, acc, false, false);
        }
        const float bb = b1[e * HH + n0 + sub];
#pragma unroll
        for (int r = 0; r < 8; ++r) {
            float v = acc[r] + bb;                       // (M = r+8*hi, N = n0+sub)
            Hs[(r + 8 * hi) * HS_STRIDE + n0 + sub] = v > 0.0f ? v : 0.0f;
        }
    }
    __syncthreads();

    // ---- GEMM2: Y = H @ W2e + b2, weighted scatter-add --------------------
#pragma unroll
    for (int nt = 0; nt < 2; ++nt) {
        const int n0 = (wv * 2 + nt) * 16;
        v8f acc = {};
        for (int k0 = 0; k0 < HH; k0 += 4) {
            const int ka = k0 + 2 * hi;
            v2f a, b;
            a.x = Hs[sub * HS_STRIDE + ka];
            a.y = Hs[sub * HS_STRIDE + ka + 1];
            b.x = W2e[(size_t)ka * DM + n0 + sub];
            b.y = W2e[(size_t)(ka + 1) * DM + n0 + sub];
            acc = __builtin_amdgcn_wmma_f32_16x16x4_f32(
                false, a, false, b, (short)0, acc, false, false);
        }
        const float bb = b2[e * DM + n0 + sub];
#pragma unroll
        for (int r = 0; r < 8; ++r) {
            const int M = r + 8 * hi;
            float val = wts[M] * (acc[r] + bb);          // padded rows: wts==0
            atomicAdd(&out[(size_t)toks[M] * DM + n0 + sub], val);
        }
    }
}

// ---------------------------------------------------------------------------
// Launcher
// inputs: x, Wg, bg, bias, W1, b1, W2, b2  (all fp32)
// ws: [counts: 8 i32 | pad to 256B | tokL: E*T i32 | wtL: E*T f32]  (~262 KB)
// ---------------------------------------------------------------------------
extern "C" void kernel_launch(void* const* d_in, const int* in_sizes, int n_in,
                              void* d_out, int out_size, void* d_ws, size_t ws_size,
                              hipStream_t stream) {
    const float* x    = (const float*)d_in[0];
    const float* Wg   = (const float*)d_in[1];
    const float* bg   = (const float*)d_in[2];
    const float* bias = (const float*)d_in[3];
    const float* W1   = (const float*)d_in[4];
    const float* b1   = (const float*)d_in[5];
    const float* W2   = (const float*)d_in[6];
    const float* b2   = (const float*)d_in[7];
    float* out = (float*)d_out;

    int*   counts = (int*)d_ws;
    int*   tokL   = (int*)((char*)d_ws + 256);
    float* wtL    = (float*)((char*)d_ws + 256 + (size_t)NE * T_TOK * sizeof(int));

    const int n_out = T_TOK * DM;
    moe_zero<<<(n_out + 255) / 256, 256, 0, stream>>>(out, n_out, counts);
    moe_gate<<<T_TOK / 4, 128, 0, stream>>>(x, Wg, bg, bias, counts, tokL, wtL);
    moe_ffn<<<dim3(T_TOK / 16, NE), 256, 0, stream>>>(
        x, W1, b1, W2, b2, counts, tokL, wtL, out);
}